// TopDecoder_55293408969103
// MI455X (gfx1250) — compile-verified
//
#include <hip/hip_runtime.h>

// ---------------------------------------------------------------------------
// TopDecoder forward for MI455X (gfx1250, wave32).
// All three GEMMs run on V_WMMA_F32_16X16X4_F32 (full f32 precision: the
// adjacency threshold at ln4 is sensitive to sub-1e-3 perturbations, so bf16
// inputs are not acceptable).  Decode (X @ X^T, 48 GFLOP) dominates and is
// compute-bound; it now uses GLOBAL_LOAD_ASYNC_TO_LDS_B32 with a double-
// buffered LDS pipeline (ASYNCcnt-gated) to overlap HBM/L2 traffic with the
// WMMA stream.  adj (64MB) stays resident in the 192MB L2 for the later
// passes.  p_norm is computed algebraically: sum(L_sys @ X) = sum_j rx[j]
//   - sum_{i,j} adj[i][j]*rx[j]/rowsum[i], avoiding a 4096x1433 matmul.
// ---------------------------------------------------------------------------

typedef __attribute__((ext_vector_type(2))) float v2f;
typedef __attribute__((ext_vector_type(8))) float v8f;

#define N_ROWS 4096
#define D_IN   1433
#define E_DIM  128
#define H_DIM  128
#define C_DIM  7
#define KC     32            // K-chunk staged in LDS
#define LDSS   34            // LDS row stride (pad: even, avoids bank conflicts)
#define LN4    1.3862943611198906f   // sigmoid(d) < 0.8  <=>  d < ln(4)

// ---- shared WMMA tile core: 8 waves, block tile 128x128, wave tile 32x64 ----
// LDS layout: As[row][k] and Bs[col][k], K-minor, stride LDSS.
// f32 16x16x4 fragment layout (ISA 7.12.2): lane 0-15 -> M/N index,
// lane>=16 selects K+2, VGPR j selects K+j  => per-fragment = one b64 load.
__device__ __forceinline__ void wmma_tile_chunk(const float* As, const float* Bs,
                                                int rowOff, int colOff, int lane,
                                                v8f acc[2][4]) {
  const int l15 = lane & 15;
  const int hi2 = (lane >> 4) << 1;
#pragma unroll
  for (int kk = 0; kk < KC; kk += 4) {
    v2f a[2], b[4];
#pragma unroll
    for (int tm = 0; tm < 2; ++tm)
      a[tm] = *(const v2f*)&As[(rowOff + tm * 16 + l15) * LDSS + kk + hi2];
#pragma unroll
    for (int tn = 0; tn < 4; ++tn)
      b[tn] = *(const v2f*)&Bs[(colOff + tn * 16 + l15) * LDSS + kk + hi2];
#pragma unroll
    for (int tm = 0; tm < 2; ++tm)
#pragma unroll
      for (int tn = 0; tn < 4; ++tn)
        acc[tm][tn] = __builtin_amdgcn_wmma_f32_16x16x4_f32(
            false, a[tm], false, b[tn], (short)0, acc[tm][tn], false, false);
  }
}

// Synchronous guarded loader (used for the K tail, zero-fills k >= K).
__device__ __forceinline__ void load_rows(float* dst, const float* src, int rowBase,
                                          int kBase, int K, int ld, int tid) {
  for (int f = tid; f < 128 * KC; f += 256) {
    int r = f >> 5, k = f & (KC - 1);
    int gk = kBase + k;
    dst[r * LDSS + k] = (gk < K) ? src[(size_t)(rowBase + r) * ld + gk] : 0.0f;
  }
}

// Async global->LDS stage of one full 128xKC tile (b32: X rows have odd float
// stride 1433, so only 4-byte alignment is guaranteed).  16 async instructions
// per thread-slice => 16 per wave for the tile (each covers 32 lanes x 4B).
// LDS VDST operand = low 32 bits of the flat shared-aperture address
// (ISA 10.2: LDS aperture truncates addr[31:0] to the LDS byte address).
__device__ __forceinline__ void async_tile_rows(float* dstLds,
                                                const float* __restrict__ src,
                                                int rowBase, int kBase, int ld,
                                                int tid) {
  for (int f = tid; f < 128 * KC; f += 256) {
    int r = f >> 5, k = f & (KC - 1);
    unsigned lds = (unsigned)(size_t)(&dstLds[r * LDSS + k]);
    const float* g = &src[(size_t)(rowBase + r) * ld + kBase + k];
    asm volatile("global_load_async_to_lds_b32 %0, %1, off"
                 :: "v"(lds), "v"(g)
                 : "memory");
  }
}

// Load 128 cols x KC k-values from a [K x ld] row-major matrix, transposed
// into LDS [col][k] (coalesced global reads, conflict-free LDS writes).
__device__ __forceinline__ void load_cols_T(float* dst, const float* src, int colBase,
                                            int kBase, int K, int ld, int tid) {
  for (int f = tid; f < 128 * KC; f += 256) {
    int k = f >> 7, c = f & 127;
    int gk = kBase + k;
    dst[c * LDSS + k] = (gk < K) ? src[(size_t)gk * ld + colBase + c] : 0.0f;
  }
}

__device__ __forceinline__ float block_reduce_sum(float v) {
  __shared__ float red[256];
  red[threadIdx.x] = v;
  __syncthreads();
  for (int s = 128; s > 0; s >>= 1) {
    if (threadIdx.x < s) red[threadIdx.x] += red[threadIdx.x + s];
    __syncthreads();
  }
  return red[0];
}

// ---- decode = X @ X^T -> threshold -> adj; fused rowsum accumulation -------
// Double-buffered: async-issue chunk c+1 while computing chunk c.
// ASYNCcnt gate: per wave each chunk = 32 async ops (16 A + 16 B); async loads
// complete in order, so waiting to <=32 outstanding guarantees chunk c landed
// while chunk c+1 stays in flight.
__global__ __launch_bounds__(256) void decode_adj_kernel(
    const float* __restrict__ X, float* __restrict__ adj,
    float* __restrict__ rowsum) {
  __shared__ float As[2][128 * LDSS];
  __shared__ float Bs[2][128 * LDSS];
  const int tid = threadIdx.x;
  const int lane = tid & 31, wave = tid >> 5;
  const int rowOff = (wave >> 1) * 32, colOff = (wave & 1) * 64;
  const int rowBase = blockIdx.y * 128, colBase = blockIdx.x * 128;
  v8f z = {};
  v8f acc[2][4];
#pragma unroll
  for (int tm = 0; tm < 2; ++tm)
#pragma unroll
    for (int tn = 0; tn < 4; ++tn) acc[tm][tn] = z;

  const int nfull = D_IN / KC;  // 44 full chunks; tail of 25 handled sync'ly

  async_tile_rows(As[0], X, rowBase, 0, D_IN, tid);
  async_tile_rows(Bs[0], X, colBase, 0, D_IN, tid);

  for (int c = 0; c < nfull; ++c) {
    if (c + 1 < nfull) {
      const int kb = (c + 1) * KC;
      async_tile_rows(As[(c + 1) & 1], X, rowBase, kb, D_IN, tid);
      async_tile_rows(Bs[(c + 1) & 1], X, colBase, kb, D_IN, tid);
      asm volatile("s_wait_asynccnt 0x20" ::: "memory");  // chunk c done
    } else {
      asm volatile("s_wait_asynccnt 0x0" ::: "memory");   // drain
    }
    __syncthreads();  // all waves' chunk-c tiles visible
    wmma_tile_chunk(As[c & 1], Bs[c & 1], rowOff, colOff, lane, acc);
    __syncthreads();  // done reading buf[c&1] before it is re-targeted
  }

  // K tail (D_IN % KC = 25): synchronous guarded load, zero-filled.
  load_rows(As[nfull & 1], X, rowBase, nfull * KC, D_IN, D_IN, tid);
  load_rows(Bs[nfull & 1], X, colBase, nfull * KC, D_IN, D_IN, tid);
  __syncthreads();
  wmma_tile_chunk(As[nfull & 1], Bs[nfull & 1], rowOff, colOff, lane, acc);

  // C/D layout: VGPR v -> M = v + 8*(lane>=16), N = lane&15.
  const int l15 = lane & 15;
  const int hiRow = (lane >> 4) << 3;
#pragma unroll
  for (int tm = 0; tm < 2; ++tm) {
    float rs[8];
#pragma unroll
    for (int v = 0; v < 8; ++v) rs[v] = 0.0f;
#pragma unroll
    for (int tn = 0; tn < 4; ++tn) {
      int col = colBase + colOff + tn * 16 + l15;
#pragma unroll
      for (int v = 0; v < 8; ++v) {
        int row = rowBase + rowOff + tm * 16 + v + hiRow;
        float d = acc[tm][tn][v];
        float a = (d < LN4) ? 1.0f : 0.0f;   // sigmoid<0.8 -> 1 else -> 0
        if (row == col) a += 1.0f;           // + eye(N)
        adj[(size_t)row * N_ROWS + col] = a;
        rs[v] += a;
      }
    }
    // reduce across the 16 lanes that share each row (masks stay in-half)
#pragma unroll
    for (int v = 0; v < 8; ++v) {
      float s = rs[v];
      s += __shfl_xor(s, 1, 32);
      s += __shfl_xor(s, 2, 32);
      s += __shfl_xor(s, 4, 32);
      s += __shfl_xor(s, 8, 32);
      if (l15 == 0)  // integer-valued floats: atomic order-independent & exact
        atomicAdd(&rowsum[rowBase + rowOff + tm * 16 + v + hiRow], s);
    }
  }
}

// ---- h = X @ nn_W + nn_b ---------------------------------------------------
__global__ __launch_bounds__(256) void h_gemm_kernel(
    const float* __restrict__ X, const float* __restrict__ W,
    const float* __restrict__ bvec, float* __restrict__ hout) {
  __shared__ float As[128 * LDSS];
  __shared__ float Bs[128 * LDSS];
  const int tid = threadIdx.x;
  const int lane = tid & 31, wave = tid >> 5;
  const int rowOff = (wave >> 1) * 32, colOff = (wave & 1) * 64;
  const int rowBase = blockIdx.x * 128;
  v8f z = {};
  v8f acc[2][4];
#pragma unroll
  for (int tm = 0; tm < 2; ++tm)
#pragma unroll
    for (int tn = 0; tn < 4; ++tn) acc[tm][tn] = z;

  for (int kb = 0; kb < D_IN; kb += KC) {
    load_rows(As, X, rowBase, kb, D_IN, D_IN, tid);
    load_cols_T(Bs, W, 0, kb, D_IN, H_DIM, tid);
    __syncthreads();
    wmma_tile_chunk(As, Bs, rowOff, colOff, lane, acc);
    __syncthreads();
  }

  const int l15 = lane & 15;
  const int hiRow = (lane >> 4) << 3;
#pragma unroll
  for (int tm = 0; tm < 2; ++tm)
#pragma unroll
    for (int tn = 0; tn < 4; ++tn) {
      int col = colOff + tn * 16 + l15;
#pragma unroll
      for (int v = 0; v < 8; ++v) {
        int row = rowBase + rowOff + tm * 16 + v + hiRow;
        hout[(size_t)row * H_DIM + col] = acc[tm][tn][v] + bvec[col];
      }
    }
}

// ---- y = adj @ h ------------------------------------------------------------
__global__ __launch_bounds__(256) void y_gemm_kernel(
    const float* __restrict__ adj, const float* __restrict__ h,
    float* __restrict__ yout) {
  __shared__ float As[128 * LDSS];
  __shared__ float Bs[128 * LDSS];
  const int tid = threadIdx.x;
  const int lane = tid & 31, wave = tid >> 5;
  const int rowOff = (wave >> 1) * 32, colOff = (wave & 1) * 64;
  const int rowBase = blockIdx.x * 128;
  v8f z = {};
  v8f acc[2][4];
#pragma unroll
  for (int tm = 0; tm < 2; ++tm)
#pragma unroll
    for (int tn = 0; tn < 4; ++tn) acc[tm][tn] = z;

  for (int kb = 0; kb < N_ROWS; kb += KC) {
    load_rows(As, adj, rowBase, kb, N_ROWS, N_ROWS, tid);
    load_cols_T(Bs, h, 0, kb, N_ROWS, H_DIM, tid);
    __syncthreads();
    wmma_tile_chunk(As, Bs, rowOff, colOff, lane, acc);
    __syncthreads();
  }

  const int l15 = lane & 15;
  const int hiRow = (lane >> 4) << 3;
#pragma unroll
  for (int tm = 0; tm < 2; ++tm)
#pragma unroll
    for (int tn = 0; tn < 4; ++tn) {
      int col = colOff + tn * 16 + l15;
#pragma unroll
      for (int v = 0; v < 8; ++v) {
        int row = rowBase + rowOff + tm * 16 + v + hiRow;
        yout[(size_t)row * H_DIM + col] = acc[tm][tn][v];
      }
    }
}

// ---- small helpers ----------------------------------------------------------
__global__ void zero_kernel(float* p, int n) {
  int i = blockIdx.x * blockDim.x + threadIdx.x;
  if (i < n) p[i] = 0.0f;
}

__global__ void rx_kernel(const float* __restrict__ X, float* __restrict__ rx) {
  int i = blockIdx.x;
  float s = 0.0f;
  for (int k = threadIdx.x; k < D_IN; k += 256) s += X[(size_t)i * D_IN + k];
  float t = block_reduce_sum(s);
  if (threadIdx.x == 0) rx[i] = t;
}

// t[i] = (sum_j adj[i][j]*rx[j]) / rowsum[i]  (deterministic per-row value)
__global__ void trow_kernel(const float* __restrict__ adj, const float* __restrict__ rx,
                            const float* __restrict__ rowsum, float* __restrict__ tarr) {
  int i = blockIdx.x;
  float s = 0.0f;
  for (int j = threadIdx.x; j < N_ROWS; j += 256)
    s += adj[(size_t)i * N_ROWS + j] * rx[j];
  float t = block_reduce_sum(s);
  if (threadIdx.x == 0) tarr[i] = t / rowsum[i];
}

// out = relu(h - y/rowsum) @ clf_W + clf_b   (4096 x 7, tiny)
__global__ void out_kernel(const float* __restrict__ h, const float* __restrict__ y,
                           const float* __restrict__ rowsum,
                           const float* __restrict__ clfW,
                           const float* __restrict__ clfb, float* __restrict__ out) {
  __shared__ float sh[H_DIM];
  int i = blockIdx.x, t = threadIdx.x;
  float inv = 1.0f / rowsum[i];
  float v = h[(size_t)i * H_DIM + t] - y[(size_t)i * H_DIM + t] * inv;
  sh[t] = fmaxf(v, 0.0f);
  __syncthreads();
  if (t < C_DIM) {
    float s = clfb[t];
    for (int j = 0; j < H_DIM; ++j) s += sh[j] * clfW[j * C_DIM + t];
    out[(size_t)i * C_DIM + t] = s;
  }
}

// p_norm = sum_j rx[j] - sum_i t[i]   (single block, deterministic)
__global__ void pnorm_kernel(const float* __restrict__ rx,
                             const float* __restrict__ tarr, float* __restrict__ p) {
  float s = 0.0f;
  for (int j = threadIdx.x; j < N_ROWS; j += 256) s += rx[j] - tarr[j];
  float t = block_reduce_sum(s);
  if (threadIdx.x == 0) p[0] = t;
}

// ---------------------------------------------------------------------------
extern "C" void kernel_launch(void* const* d_in, const int* in_sizes, int n_in,
                              void* d_out, int out_size, void* d_ws, size_t ws_size,
                              hipStream_t stream) {
  const float* X    = (const float*)d_in[0];   // [4096,1433]
  const float* emb  = (const float*)d_in[1];   // [4096,128]
  const float* nnW  = (const float*)d_in[2];   // [1433,128]
  const float* nnb  = (const float*)d_in[3];   // [128]
  const float* clfW = (const float*)d_in[4];   // [128,7]
  const float* clfb = (const float*)d_in[5];   // [7]

  float* out   = (float*)d_out;
  float* o_emb = out;                                   // 4096*128
  float* o_adj = out + (size_t)N_ROWS * E_DIM;          // 4096*4096
  float* o_out = o_adj + (size_t)N_ROWS * N_ROWS;       // 4096*7
  float* o_p   = o_out + (size_t)N_ROWS * C_DIM;        // 1

  float* ws     = (float*)d_ws;                // needs ~4.3 MB of scratch
  float* rowsum = ws;                          // 4096
  float* rx     = ws + 4096;                   // 4096
  float* tarr   = ws + 8192;                   // 4096
  float* hbuf   = ws + 12288;                  // 4096*128
  float* ybuf   = hbuf + (size_t)N_ROWS * H_DIM;  // 4096*128

  hipMemcpyAsync(o_emb, emb, (size_t)N_ROWS * E_DIM * sizeof(float),
                 hipMemcpyDeviceToDevice, stream);

  zero_kernel<<<16, 256, 0, stream>>>(rowsum, N_ROWS);
  rx_kernel<<<N_ROWS, 256, 0, stream>>>(X, rx);
  h_gemm_kernel<<<N_ROWS / 128, 256, 0, stream>>>(X, nnW, nnb, hbuf);

  dim3 grid_dec(N_ROWS / 128, N_ROWS / 128);
  decode_adj_kernel<<<grid_dec, 256, 0, stream>>>(X, o_adj, rowsum);

  trow_kernel<<<N_ROWS, 256, 0, stream>>>(o_adj, rx, rowsum, tarr);
  y_gemm_kernel<<<N_ROWS / 128, 256, 0, stream>>>(o_adj, hbuf, ybuf);
  out_kernel<<<N_ROWS, 128, 0, stream>>>(hbuf, ybuf, rowsum, clfW, clfb, o_out);
  pnorm_kernel<<<1, 256, 0, stream>>>(rx, tarr, o_p);
}